// SimpleYOLOv1Loss_86758339379422
// MI455X (gfx1250) — compile-verified
//
#include <hip/hip_runtime.h>
#include <hip/hip_bf16.h>
#include <stdint.h>

// YOLOv1 loss on MI455X (gfx1250).
// Bandwidth-bound: 80.3MB pred read once -> ~3.6us floor at 23.3 TB/s.
// No matmul structure => WMMA inapplicable; CDNA5 async global->LDS staging
// for bboxes, coalesced dword sweep of pred, wave32 shuffle reductions.

#define G 7
#define GG 49
#define NCLS 20
#define NFEAT 25
#define PB (GG * NFEAT)      // 1225 floats of pred per image
#define INV_IMG (1.0f / 448.0f)

__global__ __launch_bounds__(256) void yolo_loss_main(
    const float* __restrict__ pred,
    const float* __restrict__ bboxes,
    float* __restrict__ ws,
    int nbox)
{
    const int b   = blockIdx.x;
    const int tid = threadIdx.x;

    __shared__ float sb[64];        // staged bboxes (nbox*5 <= 64)
    __shared__ float tgt[GG][8];    // per-cell: xo, yo, sqrt(w), sqrt(h), obj, label
    __shared__ float red[8];        // one partial per wave32

    const int nb5 = nbox * 5;

    // ---- Stage bboxes[b] into LDS via CDNA5 async global->LDS loads ----
    if (tid < nb5) {
        uint32_t lds_off = (uint32_t)(uintptr_t)(&sb[tid]);
        uint64_t gaddr   = (uint64_t)(uintptr_t)(bboxes + (size_t)b * nb5 + tid);
        asm volatile("global_load_async_to_lds_b32 %0, %1, off"
                     :: "v"(lds_off), "v"(gaddr) : "memory");
    }
    asm volatile("s_wait_asynccnt 0" ::: "memory");
    __syncthreads();

    // ---- Build per-cell target (first box mapping to a cell wins) ----
    if (tid < GG) {
        float xo = 0.f, yo = 0.f, sw = 0.f, sh = 0.f, obj = 0.f, lab = -1.f;
        for (int j = 0; j < nbox; ++j) {
            float x1 = sb[j * 5 + 0] * INV_IMG;
            float y1 = sb[j * 5 + 1] * INV_IMG;
            float x2 = sb[j * 5 + 2] * INV_IMG;
            float y2 = sb[j * 5 + 3] * INV_IMG;
            float xs = (x1 + x2) * 0.5f * (float)G;
            float ys = (y1 + y2) * 0.5f * (float)G;
            float xg = floorf(xs);
            float yg = floorf(ys);
            int cell = (int)yg * G + (int)xg;
            if (cell == tid && obj == 0.f) {   // first-write-wins
                xo  = xs - xg;
                yo  = ys - yg;
                sw  = sqrtf(x2 - x1);          // pre-sqrt target w/h
                sh  = sqrtf(y2 - y1);
                obj = 1.f;
                lab = sb[j * 5 + 4];
            }
        }
        tgt[tid][0] = xo;  tgt[tid][1] = yo;
        tgt[tid][2] = sw;  tgt[tid][3] = sh;
        tgt[tid][4] = obj; tgt[tid][5] = lab;
    }
    __syncthreads();

    // ---- Coalesced sweep; incremental (cell,f) update: no per-elem div ----
    float acc = 0.f;
    const float* pb = pred + (size_t)b * PB;
    int i    = tid;
    int cell = tid / NFEAT;            // the only division
    int f    = tid - cell * NFEAT;
#pragma unroll
    for (int k = 0; k < 5; ++k) {
        if (i < PB) {
            float p   = pb[i];
            float obj = tgt[cell][4];
            bool  isq = (f == 2) | (f == 3);
            float pe  = isq ? sqrtf(p) : p;
            float te, wgt;
            if (f < 5) {
                te  = tgt[cell][f];            // slot4 == obj covers f==4
                wgt = (f == 4) ? (0.5f + 0.5f * obj) : 5.f * obj;
            } else {
                te  = ((float)(f - 5) == tgt[cell][5]) ? 1.f : 0.f;
                wgt = obj;
            }
            float d = te - pe;
            acc += wgt * d * d;
        }
        i += 256;
        f += 256 % NFEAT;              // +6
        cell += 256 / NFEAT;           // +10
        if (f >= NFEAT) { f -= NFEAT; ++cell; }
    }

    // ---- Wave32 shuffle reduction, then 8 deterministic adds ----
#pragma unroll
    for (int o = 16; o > 0; o >>= 1) acc += __shfl_xor(acc, o, 32);
    const int wave = tid >> 5, lane = tid & 31;
    if (lane == 0) red[wave] = acc;
    __syncthreads();
    if (tid == 0) {
        float s = red[0];
#pragma unroll
        for (int w = 1; w < 8; ++w) s += red[w];
        ws[b] = s;
    }
}

__global__ __launch_bounds__(1024) void yolo_loss_reduce(
    const float* __restrict__ ws, float* __restrict__ out, int B)
{
    __shared__ float red[32];
    float acc = 0.f;
    for (int i = threadIdx.x; i < B; i += 1024) acc += ws[i];  // fixed order
#pragma unroll
    for (int o = 16; o > 0; o >>= 1) acc += __shfl_xor(acc, o, 32);
    const int wave = threadIdx.x >> 5, lane = threadIdx.x & 31;
    if (lane == 0) red[wave] = acc;
    __syncthreads();
    if (threadIdx.x == 0) {
        float s = 0.f;
#pragma unroll
        for (int w = 0; w < 32; ++w) s += red[w];
        out[0] = s / (float)B;
    }
}

extern "C" void kernel_launch(void* const* d_in, const int* in_sizes, int n_in,
                              void* d_out, int out_size, void* d_ws, size_t ws_size,
                              hipStream_t stream)
{
    const float* pred   = (const float*)d_in[0];
    const float* bboxes = (const float*)d_in[1];
    float* ws  = (float*)d_ws;
    float* out = (float*)d_out;

    const int B    = in_sizes[0] / PB;                 // pred is (B,7,7,25)
    const int nbox = (B > 0) ? in_sizes[1] / (B * 5)   // bboxes is (B,n,5)
                             : 0;

    if (B <= 0) return;
    yolo_loss_main<<<B, 256, 0, stream>>>(pred, bboxes, ws, nbox);
    yolo_loss_reduce<<<1, 1024, 0, stream>>>(ws, out, B);
}